// SphericalHarmonicEdgeAttrs_49976239456302
// MI455X (gfx1250) — compile-verified
//
#include <hip/hip_runtime.h>
#include <hip/hip_bf16.h>

typedef __attribute__((ext_vector_type(2))) float v2f;
typedef __attribute__((ext_vector_type(4))) float v4f;
typedef __attribute__((ext_vector_type(8))) float v8f;

// Change-of-basis: 16 SH outputs x 20 monomials of the unit edge vector.
// Monomial order (K): 1, x, y, z, x2, xy, xz, y2, yz, z2,
//                     x3, x2y, x2z, xy2, xyz, xz2, y3, y2z, yz2, z3
// Includes e3nn 'component' normalization sqrt(2l+1) per l-block.
__device__ __constant__ float A_MAT[16][20] = {
  //        1            x            y            z            x2           xy           xz           y2           yz           z2           x3           x2y          x2z          xy2          xyz          xz2          y3           y2z          yz2          z3
  { 1.0f,         0.f,         0.f,         0.f,         0.f,         0.f,         0.f,         0.f,         0.f,         0.f,         0.f,         0.f,         0.f,         0.f,         0.f,         0.f,         0.f,         0.f,         0.f,         0.f },
  { 0.f, 1.7320508075688772f,  0.f,         0.f,         0.f,         0.f,         0.f,         0.f,         0.f,         0.f,         0.f,         0.f,         0.f,         0.f,         0.f,         0.f,         0.f,         0.f,         0.f,         0.f },
  { 0.f,         0.f, 1.7320508075688772f,  0.f,         0.f,         0.f,         0.f,         0.f,         0.f,         0.f,         0.f,         0.f,         0.f,         0.f,         0.f,         0.f,         0.f,         0.f,         0.f,         0.f },
  { 0.f,         0.f,         0.f, 1.7320508075688772f,  0.f,         0.f,         0.f,         0.f,         0.f,         0.f,         0.f,         0.f,         0.f,         0.f,         0.f,         0.f,         0.f,         0.f,         0.f,         0.f },
  { 0.f,         0.f,         0.f,         0.f,         0.f,         0.f, 3.872983346207417f,   0.f,         0.f,         0.f,         0.f,         0.f,         0.f,         0.f,         0.f,         0.f,         0.f,         0.f,         0.f,         0.f },
  { 0.f,         0.f,         0.f,         0.f,         0.f, 3.872983346207417f,   0.f,         0.f,         0.f,         0.f,         0.f,         0.f,         0.f,         0.f,         0.f,         0.f,         0.f,         0.f,         0.f,         0.f },
  { 0.f,         0.f,         0.f,         0.f, -1.118033988749895f,  0.f,         0.f, 2.23606797749979f,    0.f, -1.118033988749895f,  0.f,         0.f,         0.f,         0.f,         0.f,         0.f,         0.f,         0.f,         0.f,         0.f },
  { 0.f,         0.f,         0.f,         0.f,         0.f,         0.f,         0.f,         0.f, 3.872983346207417f,   0.f,         0.f,         0.f,         0.f,         0.f,         0.f,         0.f,         0.f,         0.f,         0.f,         0.f },
  { 0.f,         0.f,         0.f,         0.f, -1.9364916731037085f, 0.f,         0.f,         0.f,         0.f, 1.9364916731037085f,  0.f,         0.f,         0.f,         0.f,         0.f,         0.f,         0.f,         0.f,         0.f,         0.f },
  { 0.f,         0.f,         0.f,         0.f,         0.f,         0.f,         0.f,         0.f,         0.f,         0.f, -2.091650066335189f,  0.f,         0.f,         0.f,         0.f, 6.274950199005567f,   0.f,         0.f,         0.f,         0.f },
  { 0.f,         0.f,         0.f,         0.f,         0.f,         0.f,         0.f,         0.f,         0.f,         0.f,         0.f,         0.f,         0.f,         0.f, 10.246950765959598f,  0.f,         0.f,         0.f,         0.f,         0.f },
  { 0.f,         0.f,         0.f,         0.f,         0.f,         0.f,         0.f,         0.f,         0.f,         0.f, -1.6201851746019651f, 0.f,         0.f, 6.4807406984078604f,  0.f, -1.6201851746019651f, 0.f,         0.f,         0.f,         0.f },
  { 0.f,         0.f,         0.f,         0.f,         0.f,         0.f,         0.f,         0.f,         0.f,         0.f,         0.f, -3.968626966596886f,  0.f,         0.f,         0.f,         0.f, 2.6457513110645907f,  0.f, -3.968626966596886f,  0.f },
  { 0.f,         0.f,         0.f,         0.f,         0.f,         0.f,         0.f,         0.f,         0.f,         0.f,         0.f,         0.f, -1.6201851746019651f, 0.f,         0.f,         0.f,         0.f, 6.4807406984078604f,  0.f, -1.6201851746019651f },
  { 0.f,         0.f,         0.f,         0.f,         0.f,         0.f,         0.f,         0.f,         0.f,         0.f,         0.f, -5.123475382979799f,  0.f,         0.f,         0.f,         0.f,         0.f,         0.f, 5.123475382979799f,   0.f },
  { 0.f,         0.f,         0.f,         0.f,         0.f,         0.f,         0.f,         0.f,         0.f,         0.f,         0.f,         0.f, -6.274950199005567f,  0.f,         0.f,         0.f,         0.f,         0.f,         0.f, 2.091650066335189f }
};

// One edge per lane-PAIR (lane n and n+16 both own edge n of the 16-edge tile):
//  - both halves compute the same edge vector (loads merge in cache),
//  - lo half supplies B rows K=(4k,4k+1), hi half K=(4k+2,4k+3)  -> no shuffles,
//  - D VGPR r holds SH row (r + 8*hi) of edge (tile*16 + lane%16).
__global__ __launch_bounds__(256) void sh_edge_wmma_kernel(
    const float* __restrict__ pos,
    const int*   __restrict__ eidx,
    float*       __restrict__ out,
    int E)
{
  const int  lane = threadIdx.x & 31;
  const int  m    = lane & 15;     // SH row index (A) / edge-in-tile (B, D)
  const int  hi   = lane >> 4;     // which K pair of each 4-wide chunk we carry
  const long long tile  = (long long)((blockIdx.x * blockDim.x + threadIdx.x) >> 5);
  const long long ebase = tile * 16;
  if (ebase >= (long long)E) return;              // wave-uniform: EXEC stays full

  const long long e  = ebase + m;
  const long long ec = (e < (long long)E) ? e : (long long)E - 1;

  const int s = eidx[ec];
  const int t = eidx[(size_t)E + ec];

  const float vx = pos[3 * s + 0] - pos[3 * t + 0];
  const float vy = pos[3 * s + 1] - pos[3 * t + 1];
  const float vz = pos[3 * s + 2] - pos[3 * t + 2];

  const float r   = sqrtf(vx * vx + vy * vy + vz * vz);
  const float inv = 1.0f / fmaxf(r, 1e-12f);
  const float x = vx * inv, y = vy * inv, z = vz * inv;

  // Monomials of the unit vector (each half-wave only needs 10 of them;
  // the selects below compile to v_cndmask, so no cross-lane traffic).
  const float one = 1.0f;
  const float x2 = x * x,  y2 = y * y,  z2 = z * z;
  const float xy = x * y,  xz = x * z,  yz = y * z;
  const float x3 = x2 * x, x2y = x2 * y, x2z = x2 * z;
  const float xy2 = x * y2, xyz = xy * z, xz2 = x * z2;
  const float y3 = y2 * y, y2z = y2 * z, yz2 = y * z2, z3 = z2 * z;

  // B operands: v2f element i = VGPR i.  lo: K=(4k,4k+1)  hi: K=(4k+2,4k+3)
  v2f b0 = { hi ? y   : one, hi ? z   : x   };  // K 0..3 : 1,x | y,z
  v2f b1 = { hi ? xz  : x2,  hi ? y2  : xy  };  // K 4..7 : x2,xy | xz,y2
  v2f b2 = { hi ? x3  : yz,  hi ? x2y : z2  };  // K 8..11: yz,z2 | x3,x2y
  v2f b3 = { hi ? xyz : x2z, hi ? xz2 : xy2 };  // K12..15: x2z,xy2 | xyz,xz2
  v2f b4 = { hi ? yz2 : y3,  hi ? z3  : y2z };  // K16..19: y3,y2z | yz2,z3

  // A operands: constant matrix, same half-wave K striping.
  const int c0 = 2 * hi;
  v2f a0 = { A_MAT[m][ 0 + c0], A_MAT[m][ 1 + c0] };
  v2f a1 = { A_MAT[m][ 4 + c0], A_MAT[m][ 5 + c0] };
  v2f a2 = { A_MAT[m][ 8 + c0], A_MAT[m][ 9 + c0] };
  v2f a3 = { A_MAT[m][12 + c0], A_MAT[m][13 + c0] };
  v2f a4 = { A_MAT[m][16 + c0], A_MAT[m][17 + c0] };

  // SH[16 x 16edges] = A[16x20] * M[20x16], K split 5 x 4. Full f32 precision.
  v8f acc = {};
  acc = __builtin_amdgcn_wmma_f32_16x16x4_f32(false, a0, false, b0, (short)0, acc, false, false);
  acc = __builtin_amdgcn_wmma_f32_16x16x4_f32(false, a1, false, b1, (short)0, acc, false, false);
  acc = __builtin_amdgcn_wmma_f32_16x16x4_f32(false, a2, false, b2, (short)0, acc, false, false);
  acc = __builtin_amdgcn_wmma_f32_16x16x4_f32(false, a3, false, b3, (short)0, acc, false, false);
  acc = __builtin_amdgcn_wmma_f32_16x16x4_f32(false, a4, false, b4, (short)0, acc, false, false);

  // Each lane owns 8 contiguous outputs of its edge: two aligned b128 stores.
  if (e < (long long)E) {
    float* o = out + e * 16 + hi * 8;
    v4f lo4 = { acc[0], acc[1], acc[2], acc[3] };
    v4f hi4 = { acc[4], acc[5], acc[6], acc[7] };
    *(v4f*)(o)     = lo4;
    *(v4f*)(o + 4) = hi4;
  }
}

extern "C" void kernel_launch(void* const* d_in, const int* in_sizes, int n_in,
                              void* d_out, int out_size, void* d_ws, size_t ws_size,
                              hipStream_t stream) {
  (void)n_in; (void)out_size; (void)d_ws; (void)ws_size;
  const float* pos  = (const float*)d_in[0];
  const int*   eidx = (const int*)d_in[1];
  float*       out  = (float*)d_out;

  const int E = in_sizes[1] / 2;      // edge_index is [2, E] flat
  if (E <= 0) return;

  const long long waves   = ((long long)E + 15) / 16;  // 16 edges per wave32
  const long long threads = waves * 32;
  const int       block   = 256;                        // 8 waves per block
  const unsigned  grid    = (unsigned)((threads + block - 1) / block);

  sh_edge_wmma_kernel<<<grid, block, 0, stream>>>(pos, eidx, out, E);
}